// DeformableScanning_87995289961134
// MI455X (gfx1250) — compile-verified
//
#include <hip/hip_runtime.h>
#include <stdint.h>

typedef __attribute__((ext_vector_type(2))) float v2f;
typedef __attribute__((ext_vector_type(8))) float v8f;

#define BATCH 8
#define CHN   64
#define Hh    256
#define Ww    256
#define NPIX  (Hh * Ww)   /* 65536 */
#define TILE  4096        /* elements per LDS sort tile (32 KB of u64) */

// ---------------------------------------------------------------------------
// Stage 1: build sortable u64 keys:  (monotone(float) << 32) | pixel_index
// ---------------------------------------------------------------------------
__global__ void ds_keys_kernel(const float* __restrict__ delta_t,
                               unsigned long long* __restrict__ keys) {
    int gid = blockIdx.x * blockDim.x + threadIdx.x;      // over BATCH*NPIX
    int p   = gid & (NPIX - 1);
    float ref  = (float)p * (2.0f / (float)(NPIX - 1)) - 1.0f;
    float keyf = ref + delta_t[gid];
    uint32_t u = __float_as_uint(keyf);
    u = (u & 0x80000000u) ? ~u : (u | 0x80000000u);       // order-preserving map
    keys[gid] = ((unsigned long long)u << 32) | (uint32_t)p;
}

// ---------------------------------------------------------------------------
// Stage 2a: LDS bitonic local sort, stages k = 2 .. TILE
// ---------------------------------------------------------------------------
__global__ void __launch_bounds__(1024) ds_sort_local_kernel(
        unsigned long long* __restrict__ keys) {
    __shared__ unsigned long long s[TILE];
    const int base = blockIdx.x * TILE;
    for (int t = threadIdx.x; t < TILE; t += 1024) s[t] = keys[base + t];
    __syncthreads();
    for (int k = 2; k <= TILE; k <<= 1) {
        for (int j = k >> 1; j > 0; j >>= 1) {
            for (int t = 0; t < TILE / 2; t += 1024) {
                int pp = threadIdx.x + t;
                int i  = ((pp & ~(j - 1)) << 1) | (pp & (j - 1));
                int l  = i | j;
                int ib = (base + i) & (NPIX - 1);          // in-batch index
                bool asc = ((ib & k) == 0);
                unsigned long long a = s[i], b = s[l];
                bool sw = asc ? (a > b) : (a < b);
                if (sw) { s[i] = b; s[l] = a; }
            }
            __syncthreads();
        }
    }
    for (int t = threadIdx.x; t < TILE; t += 1024) keys[base + t] = s[t];
}

// ---------------------------------------------------------------------------
// Stage 2b: global bitonic step (j >= TILE)
// ---------------------------------------------------------------------------
__global__ void ds_sort_global_kernel(unsigned long long* __restrict__ keys,
                                      int k, int j) {
    int pp = blockIdx.x * blockDim.x + threadIdx.x;       // over BATCH*NPIX/2
    int i  = ((pp & ~(j - 1)) << 1) | (pp & (j - 1));
    int l  = i | j;                                        // stays in batch: j < NPIX
    int ib = i & (NPIX - 1);
    bool asc = ((ib & k) == 0);
    unsigned long long a = keys[i], b = keys[l];
    bool sw = asc ? (a > b) : (a < b);
    if (sw) { keys[i] = b; keys[l] = a; }
}

// ---------------------------------------------------------------------------
// Stage 2c: LDS bitonic local merge, j = TILE/2 .. 1 for a given k
// ---------------------------------------------------------------------------
__global__ void __launch_bounds__(1024) ds_merge_local_kernel(
        unsigned long long* __restrict__ keys, int k) {
    __shared__ unsigned long long s[TILE];
    const int base = blockIdx.x * TILE;
    for (int t = threadIdx.x; t < TILE; t += 1024) s[t] = keys[base + t];
    __syncthreads();
    for (int j = TILE / 2; j > 0; j >>= 1) {
        for (int t = 0; t < TILE / 2; t += 1024) {
            int pp = threadIdx.x + t;
            int i  = ((pp & ~(j - 1)) << 1) | (pp & (j - 1));
            int l  = i | j;
            int ib = (base + i) & (NPIX - 1);
            bool asc = ((ib & k) == 0);
            unsigned long long a = s[i], b = s[l];
            bool sw = asc ? (a > b) : (a < b);
            if (sw) { s[i] = b; s[l] = a; }
        }
        __syncthreads();
    }
    for (int t = threadIdx.x; t < TILE; t += 1024) keys[base + t] = s[t];
}

// ---------------------------------------------------------------------------
// Stage 3: invert permutation -> rank[p] = sorted position of pixel p
// ---------------------------------------------------------------------------
__global__ void ds_rank_kernel(const unsigned long long* __restrict__ keys,
                               int* __restrict__ rank) {
    int s = blockIdx.x * blockDim.x + threadIdx.x;        // over BATCH*NPIX
    int batch = s >> 16;
    int idx   = (int)(uint32_t)(keys[s] & 0xFFFFFFFFull);
    rank[(batch << 16) + idx] = s & (NPIX - 1);
}

// ---------------------------------------------------------------------------
// Stage 4: bilinear sample 16 px x 64 ch per wave, transpose each 16x16 f32
// tile in the matrix pipe via D = T x I (4 chained V_WMMA_F32_16X16X4_F32),
// then scatter contiguous channel rows to out[b, rank[p], :].
// ---------------------------------------------------------------------------
__global__ void __launch_bounds__(256) ds_sample_kernel(
        const float* __restrict__ x, const float* __restrict__ dp,
        const int* __restrict__ rank, float* __restrict__ out) {
    const int wave   = blockIdx.x * (blockDim.x >> 5) + (threadIdx.x >> 5);
    const int lane   = threadIdx.x & 31;
    const int laneLo = lane & 15;
    const int hi     = lane >> 4;                 // 0: K=0,1 half  1: K=2,3 half

    const int tile  = wave << 4;                  // first pixel (global over B*N)
    const int batch = tile >> 16;
    const int pbase = tile & (NPIX - 1);          // in-batch pixel base of tile
    const int p     = pbase + laneLo;             // this lane's pixel (dup in hi half)

    // --- per-pixel bilinear parameters (align_corners=False, zeros padding) ---
    const int  row = p >> 8, col = p & 255;
    const float dpx = dp[((size_t)batch * 2 + 0) * NPIX + p];
    const float dpy = dp[((size_t)batch * 2 + 1) * NPIX + p];
    const float gx = fmaf((float)col, 2.0f / (float)(Ww - 1), -1.0f) + dpx;
    const float gy = fmaf((float)row, 2.0f / (float)(Hh - 1), -1.0f) + dpy;
    const float ix = (gx + 1.0f) * (0.5f * (float)Ww) - 0.5f;
    const float iy = (gy + 1.0f) * (0.5f * (float)Hh) - 0.5f;
    const float ix0f = floorf(ix), iy0f = floorf(iy);
    const float wx = ix - ix0f, wy = iy - iy0f;
    const int x0 = (int)ix0f, y0 = (int)iy0f, x1 = x0 + 1, y1 = y0 + 1;

    const int cx0 = x0 < 0 ? 0 : (x0 > Ww - 1 ? Ww - 1 : x0);
    const int cx1 = x1 < 0 ? 0 : (x1 > Ww - 1 ? Ww - 1 : x1);
    const int cy0 = y0 < 0 ? 0 : (y0 > Hh - 1 ? Hh - 1 : y0);
    const int cy1 = y1 < 0 ? 0 : (y1 > Hh - 1 ? Hh - 1 : y1);
    const float mx0 = (x0 >= 0 && x0 < Ww) ? 1.0f : 0.0f;
    const float mx1 = (x1 >= 0 && x1 < Ww) ? 1.0f : 0.0f;
    const float my0 = (y0 >= 0 && y0 < Hh) ? 1.0f : 0.0f;
    const float my1 = (y1 >= 0 && y1 < Hh) ? 1.0f : 0.0f;
    const int i00 = cy0 * Ww + cx0, i10 = cy0 * Ww + cx1;
    const int i01 = cy1 * Ww + cx0, i11 = cy1 * Ww + cx1;
    const float w00 = (1.0f - wx) * (1.0f - wy) * mx0 * my0;
    const float w10 = wx * (1.0f - wy) * mx1 * my0;
    const float w01 = (1.0f - wx) * wy * mx0 * my1;
    const float w11 = wx * wy * mx1 * my1;

    const float* __restrict__ xb = x + (size_t)batch * CHN * NPIX;

    // identity B-matrix K-slices (4x16): vgpr0<->K=2*hi, vgpr1<->K=2*hi+1
    v2f bI[4];
#pragma unroll
    for (int kap = 0; kap < 4; ++kap) {
        bI[kap].x = (laneLo == 4 * kap + 2 * hi)     ? 1.0f : 0.0f;
        bI[kap].y = (laneLo == 4 * kap + 2 * hi + 1) ? 1.0f : 0.0f;
    }

    // destination sorted positions for the 8 pixel-rows this half-wave stores
    int rk[8];
#pragma unroll
    for (int r = 0; r < 8; ++r)
        rk[r] = rank[(batch << 16) + pbase + r + 8 * hi];

#pragma unroll
    for (int g = 0; g < 4; ++g) {                 // 4 channel groups of 16
        v8f d = {};
#pragma unroll
        for (int kap = 0; kap < 4; ++kap) {
            const int c0 = g * 16 + 4 * kap + 2 * hi;
            const float* pc0 = xb + (size_t)c0 * NPIX;
            const float* pc1 = pc0 + NPIX;
            v2f a;
            a.x = w00 * pc0[i00] + w10 * pc0[i10] + w01 * pc0[i01] + w11 * pc0[i11];
            a.y = w00 * pc1[i00] + w10 * pc1[i10] + w01 * pc1[i01] + w11 * pc1[i11];
            // D += T[:,4k:4k+4] x I[4k:4k+4,:]  => lane<->vgpr transpose of T
            d = __builtin_amdgcn_wmma_f32_16x16x4_f32(
                    false, a, false, bI[kap], (short)0, d, false, false);
        }
        // D layout: vgpr r -> pixel row (r + 8*hi), lane -> channel in group
#pragma unroll
        for (int r = 0; r < 8; ++r) {
            out[((size_t)batch * NPIX + rk[r]) * CHN + g * 16 + laneLo] = d[r];
        }
    }
}

// ---------------------------------------------------------------------------
extern "C" void kernel_launch(void* const* d_in, const int* in_sizes, int n_in,
                              void* d_out, int out_size, void* d_ws, size_t ws_size,
                              hipStream_t stream) {
    const float* x  = (const float*)d_in[0];   // (8,64,256,256)
    const float* dp = (const float*)d_in[1];   // (8,2,256,256)
    const float* dt = (const float*)d_in[2];   // (8,1,256,256)
    float* out = (float*)d_out;                // (8,65536,64)

    unsigned long long* keys = (unsigned long long*)d_ws;              // 4 MB
    int* rank = (int*)((char*)d_ws + (size_t)BATCH * NPIX * 8);        // 2 MB

    const int total = BATCH * NPIX;

    ds_keys_kernel<<<total / 256, 256, 0, stream>>>(dt, keys);

    ds_sort_local_kernel<<<total / TILE, 1024, 0, stream>>>(keys);
    for (int k = TILE * 2; k <= NPIX; k <<= 1) {
        for (int j = k >> 1; j >= TILE; j >>= 1)
            ds_sort_global_kernel<<<(total / 2) / 256, 256, 0, stream>>>(keys, k, j);
        ds_merge_local_kernel<<<total / TILE, 1024, 0, stream>>>(keys, k);
    }

    ds_rank_kernel<<<total / 256, 256, 0, stream>>>(keys, rank);

    // 16 pixels per wave, 8 waves per block -> 128 pixels/block
    ds_sample_kernel<<<total / 128, 256, 0, stream>>>(x, dp, rank, out);
}